// GCNNet3_15350213116648
// MI455X (gfx1250) — compile-verified
//
#include <hip/hip_runtime.h>

typedef __attribute__((ext_vector_type(2))) float v2f;
typedef __attribute__((ext_vector_type(8))) float v8f;

#define IN_F  256
#define HID_F 64
#define OUT_F 40

// ---------------- degree / norm kernels ----------------

__global__ void k_deg_init(float* __restrict__ deg, int n) {
    int i = blockIdx.x * blockDim.x + threadIdx.x;
    if (i < n) deg[i] = 1.0f;   // self-loop contributes 1 to in-degree
}

__global__ void k_deg_count(const int* __restrict__ dst, float* __restrict__ deg, int e) {
    int i = blockIdx.x * blockDim.x + threadIdx.x;
    if (i < e) unsafeAtomicAdd(&deg[dst[i]], 1.0f);
}

__global__ void k_rsqrt_inplace(float* __restrict__ deg, int n) {
    int i = blockIdx.x * blockDim.x + threadIdx.x;
    if (i < n) deg[i] = rsqrtf(deg[i]);   // deg >= 1 always (self-loop)
}

// ---------------- WMMA f32 GEMM: C[M x NCOLS] = A[M x K] @ B[K x NCOLS] ----------------
// One wave computes a 16-row x (NT*16)-col slab of C using V_WMMA_F32_16X16X4_F32.
// A-frag layout (ISA 16x4 f32): lane holds A[row0 + lane%16][kb + 2*(lane/16) + v], v=0..1
// B-frag layout (row-striped):  vgpr v holds B[kb + 2*(lane/16) + v][t*16 + lane%16]
// C/D layout: vgpr r -> row (r + 8*(lane/16)), col = lane%16.

template <int K, int NCOLS, int NT>
__global__ void k_gemm_wmma(const float* __restrict__ A, const float* __restrict__ B,
                            float* __restrict__ C, int M) {
    const int lane = threadIdx.x & 31;
    const int wave = threadIdx.x >> 5;
    const int tile = blockIdx.x * (blockDim.x >> 5) + wave;
    const int row0 = tile * 16;
    if (row0 >= M) return;                 // wave-uniform: EXEC stays all-ones

    const int lm = lane & 15;
    const int lh = lane >> 4;
    int arow = row0 + lm;
    if (arow >= M) arow = M - 1;           // clamp (dup rows; stores are guarded)

    v8f acc[NT] = {};

    for (int kb = 0; kb < K; kb += 4) {
        const int k0 = kb + 2 * lh;
        const v2f a = *(const v2f*)&A[(size_t)arow * K + k0];
#pragma unroll
        for (int t = 0; t < NT; ++t) {
            const int col = t * 16 + lm;
            v2f b;
            if ((NCOLS % 16 == 0) || (col < NCOLS)) {
                b.x = B[(size_t)k0 * NCOLS + col];
                b.y = B[(size_t)(k0 + 1) * NCOLS + col];
            } else {
                b.x = 0.0f; b.y = 0.0f;
            }
            acc[t] = __builtin_amdgcn_wmma_f32_16x16x4_f32(
                false, a, false, b, (short)0, acc[t], false, false);
        }
    }

    if (row0 + 16 <= M) {
        // Fast path (all tiles when M % 16 == 0): no row guards -> the 8 row-stores
        // per tile share one base address with immediate offsets; column guard (if
        // any) is a single per-tile exec mask, not per element.
        float* __restrict__ cbase = &C[(size_t)(row0 + 8 * lh) * NCOLS + lm];
#pragma unroll
        for (int t = 0; t < NT; ++t) {
            if ((NCOLS % 16 == 0) || (t * 16 + lm < NCOLS)) {
#pragma unroll
                for (int r = 0; r < 8; ++r)
                    cbase[(size_t)r * NCOLS + t * 16] = acc[t][r];
            }
        }
    } else {
#pragma unroll
        for (int t = 0; t < NT; ++t) {
            const int col = t * 16 + lm;
#pragma unroll
            for (int r = 0; r < 8; ++r) {
                const int row = row0 + r + 8 * lh;
                if (row < M && ((NCOLS % 16 == 0) || (col < NCOLS)))
                    C[(size_t)row * NCOLS + col] = acc[t][r];
            }
        }
    }
}

// ---------------- aggregation kernels ----------------

// out[i][f] = h[i][f] * dinv[i]^2 + bias[f]   (self-loop term + bias, both commute with sum)
template <int F>
__global__ void k_self_init(const float* __restrict__ h, const float* __restrict__ dinv,
                            const float* __restrict__ bias, float* __restrict__ out, int n) {
    long long gid = (long long)blockIdx.x * blockDim.x + threadIdx.x;
    if (gid >= (long long)n * F) return;
    int i = (int)(gid / F);
    int f = (int)(gid % F);
    float di = dinv[i];
    out[gid] = h[gid] * di * di + bias[f];
}

// per edge: out[dst] += h[src] * dinv[src]*dinv[dst]; each thread handles 4 features
template <int F>
__global__ void k_scatter(const float* __restrict__ h, const float* __restrict__ dinv,
                          const int* __restrict__ src, const int* __restrict__ dst,
                          float* __restrict__ out, int e) {
    constexpr int CHUNKS = F / 4;
    long long gid = (long long)blockIdx.x * blockDim.x + threadIdx.x;
    if (gid >= (long long)e * CHUNKS) return;
    int ed = (int)(gid / CHUNKS);
    int c  = (int)(gid % CHUNKS);
    int s = src[ed];
    int d = dst[ed];
    float w = dinv[s] * dinv[d];
    const float4 v = *(const float4*)&h[(size_t)s * F + 4 * c];
    float* o = &out[(size_t)d * F + 4 * c];
    unsafeAtomicAdd(o + 0, v.x * w);
    unsafeAtomicAdd(o + 1, v.y * w);
    unsafeAtomicAdd(o + 2, v.z * w);
    unsafeAtomicAdd(o + 3, v.w * w);
}

__global__ void k_relu(float* __restrict__ x, long long n) {
    long long i = (long long)blockIdx.x * blockDim.x + threadIdx.x;
    if (i < n) x[i] = fmaxf(x[i], 0.0f);
}

// ---------------- launch ----------------

static inline unsigned cdiv_ll(long long a, long long b) { return (unsigned)((a + b - 1) / b); }

extern "C" void kernel_launch(void* const* d_in, const int* in_sizes, int n_in,
                              void* d_out, int out_size, void* d_ws, size_t ws_size,
                              hipStream_t stream) {
    const float* x  = (const float*)d_in[0];
    const int*   ei = (const int*)d_in[1];
    const float* W1 = (const float*)d_in[2];
    const float* b1 = (const float*)d_in[3];
    const float* W2 = (const float*)d_in[4];
    const float* b2 = (const float*)d_in[5];

    const int N = in_sizes[0] / IN_F;
    const int E = in_sizes[1] / 2;
    const int* src = ei;
    const int* dst = ei + E;

    float* ws   = (float*)d_ws;
    float* dinv = ws;                         // [N]    degree -> rsqrt(degree) in place
    float* h1   = ws + N;                     // [N*64] x @ W1
    float* agg1 = h1 + (size_t)N * HID_F;     // [N*64] layer-1 aggregate / ReLU output
    float* h2   = h1;                         // [N*40] aliases h1 (dead after scatter1)
    float* out  = (float*)d_out;              // [N*40]

    const int T = 256;

    // degrees -> dinv
    k_deg_init<<<cdiv_ll(N, T), T, 0, stream>>>(dinv, N);
    k_deg_count<<<cdiv_ll(E, T), T, 0, stream>>>(dst, dinv, E);
    k_rsqrt_inplace<<<cdiv_ll(N, T), T, 0, stream>>>(dinv, N);

    // layer 1: h1 = x @ W1 (WMMA f32), then normalized scatter-sum + b1, ReLU
    {
        const int tiles = (N + 15) / 16;        // 16 rows per wave
        const int waves_per_blk = 4;            // blockDim = 128 = 4 waves
        k_gemm_wmma<IN_F, HID_F, HID_F / 16>
            <<<cdiv_ll(tiles, waves_per_blk), 32 * waves_per_blk, 0, stream>>>(x, W1, h1, N);
    }
    k_self_init<HID_F><<<cdiv_ll((long long)N * HID_F, T), T, 0, stream>>>(h1, dinv, b1, agg1, N);
    k_scatter<HID_F><<<cdiv_ll((long long)E * (HID_F / 4), T), T, 0, stream>>>(h1, dinv, src, dst, agg1, E);
    k_relu<<<cdiv_ll((long long)N * HID_F, T), T, 0, stream>>>(agg1, (long long)N * HID_F);

    // layer 2: h2 = relu(agg1) @ W2 (40 cols -> 3 guarded 16-col tiles), scatter + b2 into d_out
    {
        const int tiles = (N + 15) / 16;
        const int waves_per_blk = 4;
        k_gemm_wmma<HID_F, OUT_F, 3>
            <<<cdiv_ll(tiles, waves_per_blk), 32 * waves_per_blk, 0, stream>>>(agg1, W2, h2, N);
    }
    k_self_init<OUT_F><<<cdiv_ll((long long)N * OUT_F, T), T, 0, stream>>>(h2, dinv, b2, out, N);
    k_scatter<OUT_F><<<cdiv_ll((long long)E * (OUT_F / 4), T), T, 0, stream>>>(h2, dinv, src, dst, out, E);
}